// BinaryDiceLoss_blobPunish_90099823935613
// MI455X (gfx1250) — compile-verified
//
#include <hip/hip_runtime.h>
#include <stdint.h>

// ---------------- problem constants (from reference) ----------------
#define BATCH 16
#define HDIM  512
#define WDIM  512
#define HW    (HDIM * WDIM)            // 262144
#define NTOT  (BATCH * HW)             // 4194304

// ---------------- CC time-tiling config ----------------
#define TILE  64
#define HALO  12
#define BDIM  (TILE + 2 * HALO)        // 88
#define BAREA (BDIM * BDIM)            // 7744
#define CHUNKS (BDIM / 4)              // 22 x b128 per row
#define TPR   (WDIM / TILE)            // 8 tiles per row
#define TILES_PER_IMG (TPR * TPR)      // 64
#define NBLK_PROP (TILES_PER_IMG * BATCH) // 1024 workgroups
#define BMWORDS (NTOT / 32)            // 131072 bitmap words (512 KB)

typedef float v2f __attribute__((ext_vector_type(2)));
typedef float v8f __attribute__((ext_vector_type(8)));
typedef int   v2i __attribute__((ext_vector_type(2)));   // 8-byte aligned

// Flat (generic) pointers to LDS carry the LDS byte offset in their low 32
// bits (CDNA5 ISA 10.2); async-to-LDS ops want that.
__device__ __forceinline__ unsigned lds_addr(const void* p) {
  return (unsigned)(uintptr_t)p;
}

// ---------------- 1) global max of predict ----------------
// predict ~ U(0,1) >= 0, so IEEE order == unsigned-int order of the bits.
__global__ __launch_bounds__(256) void max_kernel(const float* __restrict__ p,
                                                  unsigned* __restrict__ maxbits,
                                                  int n) {
  __shared__ unsigned sm[256];
  unsigned m = 0u;
  for (int i = blockIdx.x * 256 + threadIdx.x; i < n; i += gridDim.x * 256)
    m = max(m, __float_as_uint(p[i]));
  sm[threadIdx.x] = m;
  __syncthreads();
  for (int o = 128; o > 0; o >>= 1) {
    if ((int)threadIdx.x < o) sm[threadIdx.x] = max(sm[threadIdx.x], sm[threadIdx.x + o]);
    __syncthreads();
  }
  if (threadIdx.x == 0) atomicMax(maxbits, sm[0]);
}

// ---------------- 2) init labels: idx+1 where mask, 0 else ----------------
__global__ __launch_bounds__(256) void init_kernel(const float* __restrict__ p,
                                                   const unsigned* __restrict__ maxbits,
                                                   int* __restrict__ labels, int n) {
  float thr = __uint_as_float(*maxbits) * 0.5f;
  int i = blockIdx.x * 256 + threadIdx.x;
  if (i < n) labels[i] = (p[i] > thr) ? (i + 1) : 0;
}

// ---------------- 3) per-batch dice sums, WMMA wave reduction ----------------
// One block per batch image. Each wave reduces its 32 partial (accN, accD)
// pairs with a single V_WMMA_F32_16X16X4_F32: A holds the pairs, B is a 0/1
// selector so column N=0 carries the accN wave-sum and N=8 carries accD.
__global__ __launch_bounds__(1024) void dice_kernel(const float* __restrict__ p,
                                                    const float* __restrict__ t,
                                                    float* __restrict__ partials) {
  __shared__ float red[64];
  const int b = blockIdx.x;
  const float* pb = p + (size_t)b * HW;
  const float* tb = t + (size_t)b * HW;
  float accN = 0.f, accD = 0.f;
  for (int i = threadIdx.x; i < HW; i += 1024) {
    float x = pb[i], y = tb[i];
    accN = fmaf(x, y, accN);
    accD = fmaf(x, x, fmaf(y, y, accD));
  }
  unsigned lane = threadIdx.x & 31u;
  unsigned ncol = lane & 15u;
  v2f a;  a.x = accN;               a.y = accD;
  v2f bm; bm.x = (ncol < 8u) ? 1.f : 0.f;  bm.y = (ncol < 8u) ? 0.f : 1.f;
  v8f c = {};
  // D[m][n<8] = accN(m)+accN(m+16), D[m][n>=8] = accD(m)+accD(m+16)
  c = __builtin_amdgcn_wmma_f32_16x16x4_f32(false, a, false, bm, (short)0, c,
                                            false, false);
  float s = c[0] + c[1] + c[2] + c[3] + c[4] + c[5] + c[6] + c[7];
  // column N=0 lives on lanes {0 (M0-7), 16 (M8-15)}; column N=8 on {8, 24}.
  float wN = __shfl(s, 0, 32) + __shfl(s, 16, 32);
  float wD = __shfl(s, 8, 32) + __shfl(s, 24, 32);
  unsigned wid = threadIdx.x >> 5;
  if (lane == 0) { red[2 * wid] = wN; red[2 * wid + 1] = wD; }
  __syncthreads();
  if (threadIdx.x == 0) {
    float sN = 0.f, sD = 0.f;
    for (int w = 0; w < 32; ++w) { sN += red[2 * w]; sD += red[2 * w + 1]; }
    partials[2 * b] = sN;
    partials[2 * b + 1] = sD;
  }
}

// ---------------- 4) time-tiled 3x3 max-pool label propagation ----------------
// Loads an 88x88 halo tile with the CDNA5 async global->LDS engine, runs k
// (<= HALO) pool iterations fully in LDS (4-pixel register blocking with
// ds_load_b64), drains the 64x64 interior back with async LDS->global stores.
// Interior pixels sit at distance >= HALO from the loaded border, so by the
// halo-distance induction they are exact after k iterations; all over-reach
// of the blocked windows lands only in garbage-tolerated border cells.
__global__ __launch_bounds__(256) void propagate_kernel(const int* __restrict__ src,
                                                        int* __restrict__ dst,
                                                        int k) {
  __shared__ int buf[2][BAREA];      // single object -> provably LDS (ds ops)
  const int tid = threadIdx.x;
  const int b  = blockIdx.x / TILES_PER_IMG;
  const int tt = blockIdx.x % TILES_PER_IMG;
  const int ty = tt / TPR, tx = tt % TPR;
  const int oy = ty * TILE - HALO;
  const int ox = tx * TILE - HALO;
  const int* sbase = src + (size_t)b * HW;

  // ---- fill: async global -> LDS (b128 fast path, b32/zero on edges) ----
  for (int i = tid; i < BDIM * CHUNKS; i += 256) {
    int r  = i / CHUNKS;
    int c4 = (i % CHUNKS) * 4;
    int gy = oy + r;
    int gx = ox + c4;
    unsigned lbase = lds_addr(&buf[0][r * BDIM + c4]);   // 16B aligned
    if (gy >= 0 && gy < HDIM && gx >= 0 && gx + 3 < WDIM) {
      unsigned goff = (unsigned)((gy * WDIM + gx) * 4);  // 16B aligned
      asm volatile("global_load_async_to_lds_b128 %0, %1, %2"
                   :: "v"(lbase), "v"(goff), "s"(sbase) : "memory");
    } else if (gy >= 0 && gy < HDIM) {
      for (int j = 0; j < 4; ++j) {
        int gxj = gx + j;
        if (gxj >= 0 && gxj < WDIM) {
          unsigned goff = (unsigned)((gy * WDIM + gxj) * 4);
          unsigned la = lbase + 4u * (unsigned)j;
          asm volatile("global_load_async_to_lds_b32 %0, %1, %2"
                       :: "v"(la), "v"(goff), "s"(sbase) : "memory");
        } else {
          buf[0][r * BDIM + c4 + j] = 0;  // maxpool pad: 0 == identity here
        }
      }
    } else {
      buf[0][r * BDIM + c4 + 0] = 0; buf[0][r * BDIM + c4 + 1] = 0;
      buf[0][r * BDIM + c4 + 2] = 0; buf[0][r * BDIM + c4 + 3] = 0;
    }
  }
  asm volatile("s_wait_asynccnt 0" ::: "memory");
  __syncthreads();

  // ---- k iterations of: new = (cur>0) ? max3x3(cur) : 0 ----
  // Rows 1..86, 22 groups of 4 columns (c0 = 1+4g). Window start index
  // e = r*88 + 4g is always even -> all v2i loads are 8-byte aligned.
  int cp = 0;
  const int NGRP = (BDIM - 2) * CHUNKS;   // 86 * 22
  for (int it = 0; it < k; ++it) {
    const int* pl = &buf[cp][0];
    int*       wp = &buf[cp ^ 1][0];
    for (int i = tid; i < NGRP; i += 256) {
      int r = i / CHUNKS + 1;             // 1..86
      int g = i % CHUNKS;                 // 0..21
      int e = r * BDIM + 4 * g;           // window cols 4g .. 4g+5 (even base)
      v2i a0 = *(const v2i*)(pl + e - BDIM);
      v2i a1 = *(const v2i*)(pl + e - BDIM + 2);
      v2i a2 = *(const v2i*)(pl + e - BDIM + 4);
      v2i b0 = *(const v2i*)(pl + e);
      v2i b1 = *(const v2i*)(pl + e + 2);
      v2i b2 = *(const v2i*)(pl + e + 4);
      v2i d0 = *(const v2i*)(pl + e + BDIM);
      v2i d1 = *(const v2i*)(pl + e + BDIM + 2);
      v2i d2 = *(const v2i*)(pl + e + BDIM + 4);
      int col0 = max(max(a0.x, b0.x), d0.x);
      int col1 = max(max(a0.y, b0.y), d0.y);
      int col2 = max(max(a1.x, b1.x), d1.x);
      int col3 = max(max(a1.y, b1.y), d1.y);
      int col4 = max(max(a2.x, b2.x), d2.x);
      int col5 = max(max(a2.y, b2.y), d2.y);
      int p12 = max(col1, col2);
      int p34 = max(col3, col4);
      int o0 = max(col0, p12);
      int o1 = max(p12, col3);
      int o2 = max(col2, p34);
      int o3 = max(p34, col5);
      wp[e + 1] = (b0.y > 0) ? o0 : 0;
      wp[e + 2] = (b1.x > 0) ? o1 : 0;
      wp[e + 3] = (b1.y > 0) ? o2 : 0;
      wp[e + 4] = (b2.x > 0) ? o3 : 0;
    }
    __syncthreads();
    cp ^= 1;
  }

  // ---- drain: async LDS -> global, 64x64 interior ----
  const int* dbase = dst + (size_t)b * HW;
  for (int i = tid; i < TILE * (TILE / 4); i += 256) {
    int r  = i / (TILE / 4);
    int c4 = (i % (TILE / 4)) * 4;
    unsigned la   = lds_addr(&buf[cp][(r + HALO) * BDIM + (c4 + HALO)]); // 16B aligned
    unsigned goff = (unsigned)(((ty * TILE + r) * WDIM + tx * TILE + c4) * 4);
    asm volatile("global_store_async_from_lds_b128 %0, %1, %2"
                 :: "v"(goff), "v"(la), "s"(dbase) : "memory");
  }
  asm volatile("s_wait_asynccnt 0" ::: "memory");
}

// ---------------- 5) unique-count bitmap ----------------
// reference label = ours-1 (foreground) or 0 (background); both land in
// [0, NTOT) -> 1 bit each. popcount(bitmap) == numel(unique(labels)).
__global__ __launch_bounds__(256) void bitmap_kernel(const int* __restrict__ labels,
                                                     unsigned* __restrict__ bitmap,
                                                     int n) {
  int i = blockIdx.x * 256 + threadIdx.x;
  if (i < n) {
    int v = labels[i];
    unsigned rv = (v > 0) ? (unsigned)(v - 1) : 0u;
    atomicOr(&bitmap[rv >> 5], 1u << (rv & 31u));
  }
}

// ---------------- 6) final combine ----------------
__global__ __launch_bounds__(1024) void final_kernel(const unsigned* __restrict__ bitmap,
                                                     const float* __restrict__ partials,
                                                     float* __restrict__ out) {
  __shared__ unsigned cnt[1024];
  unsigned c = 0u;
  for (int i = threadIdx.x; i < BMWORDS; i += 1024) c += (unsigned)__popc(bitmap[i]);
  cnt[threadIdx.x] = c;
  __syncthreads();
  for (int o = 512; o > 0; o >>= 1) {
    if ((int)threadIdx.x < o) cnt[threadIdx.x] += cnt[threadIdx.x + o];
    __syncthreads();
  }
  if (threadIdx.x == 0) {
    float nu = (float)cnt[0];
    float penalty = nu / (float)BATCH;
    penalty = (penalty < 1.0f) ? (float)BATCH : penalty;
    penalty = fminf(penalty, (float)BATCH);
    float lsum = 0.f;
    for (int b = 0; b < BATCH; ++b) {
      float num = partials[2 * b] + 1.0f;       // + SMOOTH
      float den = partials[2 * b + 1] + 1.0f;   // + SMOOTH
      lsum += 1.0f - num / den;
    }
    out[0] = (lsum / (float)BATCH) * penalty;
  }
}

// ---------------- host-side orchestration ----------------
extern "C" void kernel_launch(void* const* d_in, const int* in_sizes, int n_in,
                              void* d_out, int out_size, void* d_ws, size_t ws_size,
                              hipStream_t stream) {
  (void)in_sizes; (void)n_in; (void)out_size; (void)ws_size;
  const float* predict = (const float*)d_in[0];
  const float* target  = (const float*)d_in[1];
  float* out = (float*)d_out;

  char* ws = (char*)d_ws;
  int*      labA     = (int*)ws;                                   // 16 MB
  int*      labB     = (int*)(ws + (size_t)NTOT * 4);              // 16 MB
  unsigned* bitmap   = (unsigned*)(ws + (size_t)NTOT * 8);         // 512 KB
  float*    partials = (float*)(ws + (size_t)NTOT * 8 + BMWORDS * 4);
  unsigned* maxbits  = (unsigned*)(ws + (size_t)NTOT * 8 + BMWORDS * 4 + 256);

  hipMemsetAsync(maxbits, 0, 4, stream);
  hipMemsetAsync(bitmap, 0, (size_t)BMWORDS * 4, stream);

  max_kernel<<<1024, 256, 0, stream>>>(predict, maxbits, NTOT);
  init_kernel<<<NTOT / 256, 256, 0, stream>>>(predict, maxbits, labA, NTOT);
  dice_kernel<<<BATCH, 1024, 0, stream>>>(predict, target, partials);

  // 200 iterations = 16 passes x 12 + 1 pass x 8, ping-ponging labA/labB.
  int* s = labA; int* d = labB;
  for (int pass = 0; pass < 17; ++pass) {
    int k = (pass < 16) ? 12 : 8;
    propagate_kernel<<<NBLK_PROP, 256, 0, stream>>>(s, d, k);
    int* tmp = s; s = d; d = tmp;
  }
  // after 17 passes the current result is in `s`
  bitmap_kernel<<<NTOT / 256, 256, 0, stream>>>(s, bitmap, NTOT);
  final_kernel<<<1, 1024, 0, stream>>>(bitmap, partials, out);
}